// PureGraphConv_66340064854627
// MI455X (gfx1250) — compile-verified
//
#include <hip/hip_runtime.h>
#include <hip/hip_bf16.h>

typedef float v2f __attribute__((ext_vector_type(2)));
typedef float v8f __attribute__((ext_vector_type(8)));

// ---------- Phase 1: degree ----------
__global__ void k_init_deg(float* __restrict__ deg, int n) {
    int i = blockIdx.x * blockDim.x + threadIdx.x;
    if (i < n) deg[i] = 1.0f;  // self loop counts once
}

__global__ void k_deg_scatter(const int* __restrict__ dst, float* __restrict__ deg, int E) {
    int e = blockIdx.x * blockDim.x + threadIdx.x;
    if (e < E) atomicAdd(&deg[dst[e]], 1.0f);
}

__global__ void k_rsqrt_deg(float* __restrict__ deg, int n) {
    int i = blockIdx.x * blockDim.x + threadIdx.x;
    if (i < n) deg[i] = rsqrtf(deg[i]);  // deg >= 1 always (self loops)
}

// ---------- Phase 2: self-loop init of aggregation buffer ----------
__global__ void k_init_agg(const float* __restrict__ x, const float* __restrict__ dis,
                           float* __restrict__ agg, long total) {
    long i = (long)blockIdx.x * blockDim.x + threadIdx.x;
    if (i < total) {
        long node = i >> 6;
        float d = dis[node];
        agg[i] = x[i] * d * d;
    }
}

// ---------- Phase 3: edge gather-scale-scatter (one edge per wave32) ----------
__global__ void k_edge_scatter(const float* __restrict__ x, const int* __restrict__ src,
                               const int* __restrict__ dst, const float* __restrict__ dis,
                               float* __restrict__ agg, int E) {
    int wave = (int)(((long)blockIdx.x * blockDim.x + threadIdx.x) >> 5);
    int lane = threadIdx.x & 31;
    if (wave >= E) return;
    int s = src[wave];
    int d = dst[wave];
    float norm = dis[s] * dis[d];
    const float2* xs = (const float2*)(x + (size_t)s * 64);
    float2 v = xs[lane];                       // 32 lanes x 8B = full 256B row, coalesced
    float* ap = agg + (size_t)d * 64 + lane * 2;
    atomicAdd(ap,     v.x * norm);
    atomicAdd(ap + 1, v.y * norm);
}

// ---------- Phase 4: out = agg @ W^T + b via V_WMMA_F32_16X16X4_F32 ----------
__global__ void __launch_bounds__(256) k_gemm_wmma(const float* __restrict__ agg,
                                                   const float* __restrict__ W,
                                                   const float* __restrict__ bias,
                                                   float* __restrict__ out, int nTilesM) {
    int wave = (int)(((long)blockIdx.x * blockDim.x + threadIdx.x) >> 5);
    if (wave >= nTilesM) return;               // wave-uniform: EXEC stays all-ones
    int lane = threadIdx.x & 31;
    int half = lane >> 4;                      // 0 or 1
    int l16  = lane & 15;
    int m0   = wave * 16;

    // A operand (16x4 f32): row = l16, K pair selected by half
    const float* arow = agg + (size_t)(m0 + l16) * 64 + 2 * half;
    // B operand (4x16 f32) = W^T tile: b[v] = W[n0*16 + l16][k + 2*half + v]
    const float* w0 = W + (size_t)(0 * 16 + l16) * 64 + 2 * half;
    const float* w1 = W + (size_t)(1 * 16 + l16) * 64 + 2 * half;
    const float* w2 = W + (size_t)(2 * 16 + l16) * 64 + 2 * half;
    const float* w3 = W + (size_t)(3 * 16 + l16) * 64 + 2 * half;

    v8f c0 = {}, c1 = {}, c2 = {}, c3 = {};
#pragma unroll
    for (int kk = 0; kk < 16; ++kk) {
        v2f a  = *(const v2f*)(arow + kk * 4);
        v2f b0 = *(const v2f*)(w0 + kk * 4);
        v2f b1 = *(const v2f*)(w1 + kk * 4);
        v2f b2 = *(const v2f*)(w2 + kk * 4);
        v2f b3 = *(const v2f*)(w3 + kk * 4);
        c0 = __builtin_amdgcn_wmma_f32_16x16x4_f32(false, a, false, b0, (short)0, c0, false, false);
        c1 = __builtin_amdgcn_wmma_f32_16x16x4_f32(false, a, false, b1, (short)0, c1, false, false);
        c2 = __builtin_amdgcn_wmma_f32_16x16x4_f32(false, a, false, b2, (short)0, c2, false, false);
        c3 = __builtin_amdgcn_wmma_f32_16x16x4_f32(false, a, false, b3, (short)0, c3, false, false);
    }

    // C/D layout: N = l16, M = vgpr + 8*half
    int rowBase = m0 + 8 * half;
    float bi0 = bias[ 0 + l16];
    float bi1 = bias[16 + l16];
    float bi2 = bias[32 + l16];
    float bi3 = bias[48 + l16];
#pragma unroll
    for (int v = 0; v < 8; ++v) {
        float* o = out + (size_t)(rowBase + v) * 64;
        o[ 0 + l16] = c0[v] + bi0;
        o[16 + l16] = c1[v] + bi1;
        o[32 + l16] = c2[v] + bi2;
        o[48 + l16] = c3[v] + bi3;
    }
}

// Scalar tail for rows not covered by 16-row tiles (unused when n % 16 == 0)
__global__ void k_gemm_tail(const float* __restrict__ agg, const float* __restrict__ W,
                            const float* __restrict__ bias, float* __restrict__ out,
                            int rowStart, int n) {
    int idx = blockIdx.x * blockDim.x + threadIdx.x;
    int nRows = n - rowStart;
    if (idx >= nRows * 64) return;
    int row = rowStart + idx / 64;
    int col = idx & 63;
    const float* a = agg + (size_t)row * 64;
    const float* w = W + (size_t)col * 64;
    float acc = bias[col];
    for (int k = 0; k < 64; ++k) acc += a[k] * w[k];
    out[(size_t)row * 64 + col] = acc;
}

extern "C" void kernel_launch(void* const* d_in, const int* in_sizes, int n_in,
                              void* d_out, int out_size, void* d_ws, size_t ws_size,
                              hipStream_t stream) {
    const float* x    = (const float*)d_in[0];
    const int*   ei   = (const int*)d_in[1];
    const float* W    = (const float*)d_in[2];
    const float* bias = (const float*)d_in[3];
    float*       out  = (float*)d_out;

    const int n = in_sizes[0] / 64;
    const int E = in_sizes[1] / 2;
    const int* src = ei;
    const int* dst = ei + E;

    // workspace: [deg/dis: n floats][agg: n*64 floats]
    float* deg = (float*)d_ws;
    size_t degBytes = (((size_t)n * sizeof(float)) + 255) & ~(size_t)255;
    float* agg = (float*)((char*)d_ws + degBytes);

    const int T = 256;
    k_init_deg   <<<(n + T - 1) / T, T, 0, stream>>>(deg, n);
    k_deg_scatter<<<(E + T - 1) / T, T, 0, stream>>>(dst, deg, E);
    k_rsqrt_deg  <<<(n + T - 1) / T, T, 0, stream>>>(deg, n);

    long total = (long)n * 64;
    k_init_agg<<<(unsigned)((total + T - 1) / T), T, 0, stream>>>(x, deg, agg, total);

    long edgeThreads = (long)E * 32;
    k_edge_scatter<<<(unsigned)((edgeThreads + T - 1) / T), T, 0, stream>>>(x, src, dst, deg, agg, E);

    int nFull = n / 16;
    if (nFull > 0)
        k_gemm_wmma<<<(nFull + 7) / 8, T, 0, stream>>>(agg, W, bias, out, nFull);
    int tailRows = n - nFull * 16;
    if (tailRows > 0)
        k_gemm_tail<<<(tailRows * 64 + T - 1) / T, T, 0, stream>>>(agg, W, bias, out, nFull * 16, n);
}